// FPNAttentionV2_35287451304849
// MI455X (gfx1250) — compile-verified
//
#include <hip/hip_runtime.h>

typedef __attribute__((ext_vector_type(2))) float v2f;
typedef __attribute__((ext_vector_type(8))) float v8f;

#define C_DIM 256

// ---------------------------------------------------------------------------
// Pointwise (1x1 conv) GEMM:  out[p][o] = sum_c x[b][c][p] * w[o][c] + b[o]
// x NCHW (per-batch plane stride HW), out NHWC flat.
// Block: 256 threads = 8 waves. Block tile: M=32 pixels x N=256 outputs.
// Wave (wid): m0 = (wid&1)*16, n0 = (wid>>1)*64  -> 4 x v8f accumulators.
// LDS tiles are stored K-fastest with even row pitch (34) so every WMMA
// A/B fragment is a single 8-byte-aligned ds_load_b64.
// upsample != 0: HW==4096 (64x64) and each result is replicated 2x2 into a
// 128x128 NHWC output (nearest upsample fused into the store).
// ---------------------------------------------------------------------------
__global__ __launch_bounds__(256) void pw_gemm_kernel(
    const float* __restrict__ x, const float* __restrict__ w,
    const float* __restrict__ bias, float* __restrict__ out,
    int HW, int upsample)
{
    __shared__ float As[32][34];     // [m][k], pitch 34 (even, conflict-free)
    __shared__ float Ws[256][34];    // [o][k], pitch 34

    const int t    = threadIdx.x;
    const int lane = t & 31;
    const int hl   = lane >> 4;      // half-wave: K offset selector
    const int row  = lane & 15;      // M (for A) / N (for B,C,D) index
    const int wid  = t >> 5;
    const int m0   = (wid & 1) * 16;
    const int n0   = (wid >> 1) * 64;

    const int p_base   = blockIdx.x * 32;          // global pixel base
    const int b        = p_base / HW;              // batch (tile never crosses)
    const int sp_base  = p_base - b * HW;
    const float* xb    = x + (size_t)b * C_DIM * HW + sp_base;

    v8f acc[4] = {};

    for (int kc = 0; kc < C_DIM; kc += 32) {
        // ---- stage A tile: As[m][k] = x[b][kc+k][sp_base+m] (coalesced) ----
        #pragma unroll
        for (int i = 0; i < 4; ++i) {
            int idx = t + 256 * i;
            int k = idx >> 5, m = idx & 31;
            As[m][k] = xb[(size_t)(kc + k) * HW + m];
            if (kc + 32 < C_DIM)                    // prefetch next K chunk
                __builtin_prefetch(&xb[(size_t)(kc + 32 + k) * HW + m], 0, 1);
        }
        // ---- stage W tile: Ws[o][k] = w[o][kc+k] (float4 row reads) ----
        {
            const float4* wr = (const float4*)(w + (size_t)t * C_DIM + kc);
            #pragma unroll
            for (int j = 0; j < 8; ++j) {
                float4 v = wr[j];
                *(v2f*)&Ws[t][4 * j + 0] = v2f{v.x, v.y};
                *(v2f*)&Ws[t][4 * j + 2] = v2f{v.z, v.w};
            }
        }
        __syncthreads();

        #pragma unroll
        for (int k4 = 0; k4 < 32; k4 += 4) {
            const v2f a = *(const v2f*)&As[m0 + row][k4 + 2 * hl];
            #pragma unroll
            for (int i = 0; i < 4; ++i) {
                const v2f bf = *(const v2f*)&Ws[n0 + i * 16 + row][k4 + 2 * hl];
                acc[i] = __builtin_amdgcn_wmma_f32_16x16x4_f32(
                    false, a, false, bf, (short)0, acc[i], false, false);
            }
        }
        __syncthreads();
    }

    // ---- epilogue: bias + NHWC store (optionally 2x2 nearest upsample) ----
    #pragma unroll
    for (int i = 0; i < 4; ++i) {
        const int col = n0 + i * 16 + row;
        const float bv = bias[col];
        #pragma unroll
        for (int j = 0; j < 8; ++j) {
            const int M   = m0 + j + 8 * hl;    // C/D layout: VGPR j -> M=j / j+8
            const float v = acc[i][j] + bv;
            const int p   = p_base + M;
            if (!upsample) {
                out[(size_t)p * C_DIM + col] = v;
            } else {
                // p indexes [2,64,64]; write to 2x2 block of [2,128,128] NHWC
                const int bb = p >> 12;
                const int sp = p & 4095;
                const int h2 = sp >> 6, w2 = sp & 63;
                const size_t ob = (size_t)bb * 16384 + (size_t)(h2 * 2) * 128 + w2 * 2;
                out[(ob      ) * C_DIM + col] = v;
                out[(ob +   1) * C_DIM + col] = v;
                out[(ob + 128) * C_DIM + col] = v;
                out[(ob + 129) * C_DIM + col] = v;
            }
        }
    }
}

// ---------------------------------------------------------------------------
// 3x3 stride-2 pad-1 conv as implicit GEMM (K = 256*9), WMMA f32 16x16x4.
// x1: [2,256,128,128]; w: [256,256,3,3]; out NHWC [2,64,64,256].
// Block tile = 32 output pixels (half an output row) x all 256 outputs.
// ---------------------------------------------------------------------------
__global__ __launch_bounds__(256) void conv3x3s2_kernel(
    const float* __restrict__ x, const float* __restrict__ w,
    const float* __restrict__ bias, float* __restrict__ out)
{
    __shared__ float As[32][34];
    __shared__ float Ws[256][34];

    const int t    = threadIdx.x;
    const int lane = t & 31;
    const int hl   = lane >> 4;
    const int row  = lane & 15;
    const int wid  = t >> 5;
    const int m0   = (wid & 1) * 16;
    const int n0   = (wid >> 1) * 64;

    const int p_base = blockIdx.x * 32;     // output pixel base
    const int b      = p_base >> 12;        // 64*64 = 4096 per batch
    const int sp     = p_base & 4095;
    const int ho     = sp >> 6;
    const int wbase  = sp & 63;             // 0 or 32
    const float* xb  = x + (size_t)b * 256 * 16384;

    v8f acc[4] = {};

    for (int kh = 0; kh < 3; ++kh) {
        const int hi = 2 * ho + kh - 1;
        if (hi < 0 || hi >= 128) continue;          // uniform per block
        for (int kw = 0; kw < 3; ++kw) {
            for (int kc = 0; kc < 256; kc += 32) {
                // A tile: stride-2 gather along W with zero padding
                #pragma unroll
                for (int i = 0; i < 4; ++i) {
                    int idx = t + 256 * i;
                    int k = idx >> 5, m = idx & 31;
                    int wi = 2 * (wbase + m) + kw - 1;
                    float v = 0.0f;
                    if (wi >= 0 && wi < 128)
                        v = xb[(size_t)(kc + k) * 16384 + hi * 128 + wi];
                    As[m][k] = v;
                }
                // W tile: Ws[o][k] = w[o][kc+k][kh][kw]  (stride-9 gather)
                {
                    const float* wr = w + (size_t)t * 2304 + kh * 3 + kw;
                    #pragma unroll
                    for (int j = 0; j < 32; ++j)
                        Ws[t][j] = wr[(size_t)(kc + j) * 9];
                }
                __syncthreads();

                #pragma unroll
                for (int k4 = 0; k4 < 32; k4 += 4) {
                    const v2f a = *(const v2f*)&As[m0 + row][k4 + 2 * hl];
                    #pragma unroll
                    for (int i = 0; i < 4; ++i) {
                        const v2f bf =
                            *(const v2f*)&Ws[n0 + i * 16 + row][k4 + 2 * hl];
                        acc[i] = __builtin_amdgcn_wmma_f32_16x16x4_f32(
                            false, a, false, bf, (short)0, acc[i], false, false);
                    }
                }
                __syncthreads();
            }
        }
    }

    #pragma unroll
    for (int i = 0; i < 4; ++i) {
        const int col = n0 + i * 16 + row;
        const float bv = bias[col];
        #pragma unroll
        for (int j = 0; j < 8; ++j) {
            const int M = m0 + j + 8 * hl;
            out[(size_t)(p_base + M) * 256 + col] = acc[i][j] + bv;
        }
    }
}

extern "C" void kernel_launch(void* const* d_in, const int* in_sizes, int n_in,
                              void* d_out, int out_size, void* d_ws, size_t ws_size,
                              hipStream_t stream) {
    const float* x1    = (const float*)d_in[0];
    const float* x2    = (const float*)d_in[1];
    const float* q1_w  = (const float*)d_in[2];
    const float* q1_b  = (const float*)d_in[3];
    const float* q2_w  = (const float*)d_in[4];
    const float* q2_b  = (const float*)d_in[5];
    const float* ks1_w = (const float*)d_in[6];
    const float* ks1_b = (const float*)d_in[7];
    const float* ks2_w = (const float*)d_in[8];
    const float* ks2_b = (const float*)d_in[9];
    const float* kup_w = (const float*)d_in[10];
    const float* kup_b = (const float*)d_in[11];
    const float* v1_w  = (const float*)d_in[12];
    const float* v1_b  = (const float*)d_in[13];
    const float* v2_w  = (const float*)d_in[14];
    const float* v2_b  = (const float*)d_in[15];
    const float* kd_w  = (const float*)d_in[16];
    const float* kd_b  = (const float*)d_in[17];

    float* out = (float*)d_out;
    const size_t N1 = (size_t)2 * 128 * 128 * 256;   // 8388608
    const size_t N2 = (size_t)2 * 64 * 64 * 256;     // 2097152
    float* o_q1  = out;
    float* o_q2  = out + N1;
    float* o_kup = out + N1 + N2;
    float* o_ks1 = out + 2 * N1 + N2;
    float* o_ks2 = out + 3 * N1 + N2;
    float* o_kd  = out + 3 * N1 + 2 * N2;
    float* o_v1  = out + 3 * N1 + 3 * N2;
    float* o_v2  = out + 4 * N1 + 3 * N2;

    dim3 blk(256);
    pw_gemm_kernel<<<1024, blk, 0, stream>>>(x1, q1_w,  q1_b,  o_q1,  16384, 0);
    pw_gemm_kernel<<<256,  blk, 0, stream>>>(x2, q2_w,  q2_b,  o_q2,  4096,  0);
    pw_gemm_kernel<<<256,  blk, 0, stream>>>(x2, kup_w, kup_b, o_kup, 4096,  1);
    pw_gemm_kernel<<<1024, blk, 0, stream>>>(x1, ks1_w, ks1_b, o_ks1, 16384, 0);
    pw_gemm_kernel<<<256,  blk, 0, stream>>>(x2, ks2_w, ks2_b, o_ks2, 4096,  0);
    conv3x3s2_kernel<<<256, blk, 0, stream>>>(x1, kd_w, kd_b, o_kd);
    pw_gemm_kernel<<<1024, blk, 0, stream>>>(x1, v1_w,  v1_b,  o_v1,  16384, 0);
    pw_gemm_kernel<<<256,  blk, 0, stream>>>(x2, v2_w,  v2_b,  o_v2,  4096,  0);
}